// RoBERTa_7000796692884
// MI455X (gfx1250) — compile-verified
//
#include <hip/hip_runtime.h>
#include <hip/hip_bf16.h>

#define BB   256
#define SS   512
#define HH   1024
#define NSEG 15
#define LDX  1040      // 1024 pooled + 15 sims + 1 zero pad (multiple of 4)
#define KW1  1039      // H + NSEG, actual K of layer 1
#define EPSF 1e-8f

typedef __attribute__((ext_vector_type(2))) float v2f;
typedef __attribute__((ext_vector_type(8))) float v8f;

// ---------------------------------------------------------------------------
// Kernel 0: zero-pad W1 [1039][128] -> W1p [1040][128] so the WMMA K-loop is
// branch-free (no per-iteration exec-mask juggling for the last row).
// ---------------------------------------------------------------------------
__global__ __launch_bounds__(256) void pad_w1_kernel(
    const float* __restrict__ W1, float* __restrict__ W1p)
{
  const int idx = blockIdx.x * 256 + threadIdx.x;   // 0 .. LDX*128-1
  if (idx < LDX * 128) {
    W1p[idx] = (idx < KW1 * 128) ? W1[idx] : 0.0f;
  }
}

// ---------------------------------------------------------------------------
// Kernel 1: per-batch title mean, segment means, cosine sims; assemble X rows.
// Grid = B blocks, 256 threads (8 waves). Thread t owns columns 4t..4t+3.
// Reads only the rows actually referenced (no cumsum materialization) —
// this is the HBM-bound part (~300 of 512 rows per batch on average).
// ---------------------------------------------------------------------------
__global__ __launch_bounds__(256) void seg_sims_kernel(
    const float* __restrict__ seq,     // [B,S,H]
    const float* __restrict__ pooled,  // [B,H]
    const int*   __restrict__ lot,     // [B,16]
    float*       __restrict__ X)       // [B,LDX] workspace
{
  const int b = blockIdx.x;
  const int t = threadIdx.x;

  __shared__ int   s_start[NSEG], s_end[NSEG], s_cnt[NSEG], s_valid[NSEG];
  __shared__ float s_sims[NSEG];
  __shared__ float s_tnorm;
  __shared__ float red0[256];
  __shared__ float red1[256];

  const int L0 = lot[b * 16 + 0];

  if (t == 0) {
    int pos = L0 + 2;
    int run = 1;
    for (int k = 0; k < NSEG; ++k) {
      const int len = lot[b * 16 + 1 + k];
      const int rs  = pos;
      pos += len;
      run &= ((len > 0) && (rs < SS)) ? 1 : 0;
      s_valid[k] = run;
      int st = rs;       if (st < 0) st = 0;   if (st > SS) st = SS;
      int en = rs + len; if (en < 0) en = 0;   if (en > SS) en = SS;
      if (en < st) en = st;
      s_start[k] = st;
      s_end[k]   = en;
      int c = en - st;   if (c < 1) c = 1;
      s_cnt[k]   = c;
    }
  }
  __syncthreads();

  const float* sb = seq + (size_t)b * SS * HH;

  // ---- title mean over rows [1, L0] ----
  float tx = 0.f, ty = 0.f, tz = 0.f, tw = 0.f;
  for (int r = 1; r <= L0; ++r) {
    const float4 v = *(const float4*)(sb + (size_t)r * HH + 4 * t);
    tx += v.x; ty += v.y; tz += v.z; tw += v.w;
  }
  const float invL = 1.0f / (float)L0;
  tx *= invL; ty *= invL; tz *= invL; tw *= invL;

  // ---- ||title|| ----
  red0[t] = tx * tx + ty * ty + tz * tz + tw * tw;
  __syncthreads();
  for (int s = 128; s >= 1; s >>= 1) {
    if (t < s) red0[t] += red0[t + s];
    __syncthreads();
  }
  if (t == 0) s_tnorm = fmaxf(sqrtf(red0[0]), EPSF);
  __syncthreads();
  const float tnorm = s_tnorm;

  // ---- segments: mean, dot(title,mean), ||mean|| -> cosine ----
  for (int k = 0; k < NSEG; ++k) {
    const int st = s_start[k];
    const int en = s_end[k];
    float sx = 0.f, sy = 0.f, sz = 0.f, sw = 0.f;
    for (int r = st; r < en; ++r) {
      const float4 v = *(const float4*)(sb + (size_t)r * HH + 4 * t);
      sx += v.x; sy += v.y; sz += v.z; sw += v.w;
    }
    const float invC = 1.0f / (float)s_cnt[k];
    sx *= invC; sy *= invC; sz *= invC; sw *= invC;

    red0[t] = tx * sx + ty * sy + tz * sz + tw * sw;   // partial dot
    red1[t] = sx * sx + sy * sy + sz * sz + sw * sw;   // partial |mean|^2
    __syncthreads();
    for (int s = 128; s >= 1; s >>= 1) {
      if (t < s) { red0[t] += red0[t + s]; red1[t] += red1[t + s]; }
      __syncthreads();
    }
    if (t == 0) {
      const float snorm = fmaxf(sqrtf(red1[0]), EPSF);
      const float cosv  = red0[0] / (tnorm * snorm);
      s_sims[k] = s_valid[k] ? cosv : 0.0f;
    }
    __syncthreads();
  }

  // ---- assemble X row: [pooled (1024) | sims (15) | 0 pad (1)] ----
  const float4 pv = *(const float4*)(pooled + (size_t)b * HH + 4 * t);
  *(float4*)(X + (size_t)b * LDX + 4 * t) = pv;
  if (t < 16) {
    X[(size_t)b * LDX + HH + t] = (t < NSEG) ? s_sims[t] : 0.0f;
  }
}

// ---------------------------------------------------------------------------
// Kernel 2: fused 3-layer MLP on the WMMA pipe (f32 16x16x4, full precision).
// Grid = 16 blocks x 16 rows; 8 waves/block, one 16x16 N-tile per wave.
//   A 16x4 f32: lanes 0-15 -> K={0,1}, lanes 16-31 -> K={2,3} (ISA 7.12.2)
//   C/D 16x16 f32: VGPR i -> M = i + 8*(lane/16), N = lane%16 (+ tile base)
// Layer-1 K-loop is branch-free: X padded to K=1040 (zero col), W1p padded.
// ---------------------------------------------------------------------------
__global__ __launch_bounds__(256) void mlp_wmma_kernel(
    const float* __restrict__ X,   // [B,LDX]
    const float* __restrict__ W1p, // [1040,128] zero-padded
    const float* __restrict__ b1,  // [128]
    const float* __restrict__ W2,  // [128,128]
    const float* __restrict__ b2,  // [128]
    const float* __restrict__ W3,  // [128,2]
    const float* __restrict__ b3,  // [2]
    float*       __restrict__ out) // [B,2]
{
  const int tid   = threadIdx.x;
  const int wave  = tid >> 5;        // 0..7 -> N tile
  const int lane  = tid & 31;
  const int row   = lane & 15;       // M (for A) / N-within-tile (for B,C,D)
  const int khalf = lane >> 4;       // 0: K={0,1}, 1: K={2,3}
  const int m0    = blockIdx.x * 16;
  const int n     = wave * 16 + row; // output column 0..127

  __shared__ float h1s[16][132];     // pad 132 -> conflict-free column reads
  __shared__ float h2s[16][132];

  // ---------- layer 1: h1 = leaky(X @ W1 + b1), K padded to 1040 ----
  v8f acc = {};
  const float* xrow = X + (size_t)(m0 + row) * LDX;
  for (int k = 0; k < LDX; k += 4) {
    const int ka = k + 2 * khalf;                 // even, <= 1038
    v2f a;  a[0] = xrow[ka];            a[1] = xrow[ka + 1];
    v2f bm; bm[0] = W1p[ka * 128 + n];  bm[1] = W1p[(ka + 1) * 128 + n];
    acc = __builtin_amdgcn_wmma_f32_16x16x4_f32(
        false, a, false, bm, (short)0, acc, false, false);
  }
  {
    const float bias = b1[n];
#pragma unroll
    for (int i = 0; i < 8; ++i) {
      float h = acc[i] + bias;
      h = (h >= 0.0f) ? h : 0.1f * h;
      h1s[i + 8 * khalf][n] = h;
    }
  }
  __syncthreads();

  // ---------- layer 2: h2 = leaky(h1 @ W2 + b2), K = 128 ----
  v8f acc2 = {};
  for (int k = 0; k < 128; k += 4) {
    const int ka = k + 2 * khalf;
    v2f a;  a[0] = h1s[row][ka];       a[1] = h1s[row][ka + 1];
    v2f bm; bm[0] = W2[ka * 128 + n];  bm[1] = W2[(ka + 1) * 128 + n];
    acc2 = __builtin_amdgcn_wmma_f32_16x16x4_f32(
        false, a, false, bm, (short)0, acc2, false, false);
  }
  {
    const float bias = b2[n];
#pragma unroll
    for (int i = 0; i < 8; ++i) {
      float h = acc2[i] + bias;
      h = (h >= 0.0f) ? h : 0.1f * h;
      h2s[i + 8 * khalf][n] = h;
    }
  }
  __syncthreads();

  // ---------- layer 3: logits = h2 @ W3 + b3 (128 -> 2), tiny: one wave ----
  if (tid < 32) {
    const int m   = tid >> 1;        // 0..15
    const int col = tid & 1;         // 0..1
    float s = b3[col];
    for (int k = 0; k < 128; ++k) s += h2s[m][k] * W3[k * 2 + col];
    out[(m0 + m) * 2 + col] = s;
  }
}

extern "C" void kernel_launch(void* const* d_in, const int* in_sizes, int n_in,
                              void* d_out, int out_size, void* d_ws, size_t ws_size,
                              hipStream_t stream) {
  const float* seq    = (const float*)d_in[0];
  const float* pooled = (const float*)d_in[1];
  const int*   lot    = (const int*)  d_in[2];
  const float* W1     = (const float*)d_in[3];
  const float* b1     = (const float*)d_in[4];
  const float* W2     = (const float*)d_in[5];
  const float* b2     = (const float*)d_in[6];
  const float* W3     = (const float*)d_in[7];
  const float* b3     = (const float*)d_in[8];
  float* out = (float*)d_out;

  float* X   = (float*)d_ws;                    // [B, LDX]   = ~1.06 MB
  float* W1p = X + (size_t)BB * LDX;            // [LDX, 128] = ~0.53 MB

  pad_w1_kernel<<<(LDX * 128 + 255) / 256, 256, 0, stream>>>(W1, W1p);
  seg_sims_kernel<<<BB, 256, 0, stream>>>(seq, pooled, lot, X);
  mlp_wmma_kernel<<<BB / 16, 256, 0, stream>>>(X, W1p, b1, W2, b2, W3, b3, out);
}